// Operator_59562606461371
// MI455X (gfx1250) — compile-verified
//
#include <hip/hip_runtime.h>

// ---------------------------------------------------------------------------
// Tri3 FEM energy integration for MI455X (gfx1250, wave32).
//   total = sum_e detJ_e * ( (1/6)*(val_q0+val_q1+val_q2) + 0.5*grad_energy_e )
// Stage 1: grid-stride gather+compute, per-block WMMA-assisted reduction -> d_ws
// Stage 2: single block reduces block partials (deterministic, no atomics).
// ---------------------------------------------------------------------------

typedef float v2f __attribute__((ext_vector_type(2)));
typedef float v8f __attribute__((ext_vector_type(8)));

// Block-wide (256-thread, 8-wave) sum using V_WMMA_F32_16X16X4_F32.
// Wave 0 feeds the 256 staged partials through B (4 chunks of 64 values) with
// A = all-ones; every row of D = column sums of B, accumulated via C. Summing
// row 0 across 16 lanes yields the exact f32 total regardless of the precise
// B-matrix lane layout (any bijection of 64 values sums identically).
__device__ __forceinline__ float block_reduce_wmma(float acc) {
    __shared__ float sdata[256];
    sdata[threadIdx.x] = acc;
    __syncthreads();

    float total = 0.0f;
    if (threadIdx.x < 32) {            // wave 0 fully active -> EXEC all ones
        const int lane = threadIdx.x;
        v2f a; a.x = 1.0f; a.y = 1.0f; // all-ones A (16x4), layout-invariant
        v8f c = {};
#pragma unroll
        for (int chunk = 0; chunk < 4; ++chunk) {
            v2f b;
            b.x = sdata[chunk * 64 + lane];
            b.y = sdata[chunk * 64 + 32 + lane];
            // D = A(ones) x B(partials) + C  -> rows of D = column sums of B
            c = __builtin_amdgcn_wmma_f32_16x16x4_f32(
                    /*neg_a=*/false, a, /*neg_b=*/false, b,
                    /*c_mod=*/(short)0, c, /*reuse_a=*/false, /*reuse_b=*/false);
        }
        float v = c[0];                // row M=0: lanes 0..15 hold 16 col sums
#pragma unroll
        for (int off = 8; off >= 1; off >>= 1)
            v += __shfl_xor(v, off, 16);
        total = v;                     // valid on lane 0
    }
    return total;
}

__global__ __launch_bounds__(256)
void fem_energy_kernel(const float* __restrict__ nodal_values,  // (N_NODES,4)
                       const float* __restrict__ coords,        // (N_NODES,2)
                       const int*   __restrict__ elements,      // (N_ELEM,3)
                       float* __restrict__ block_sums,
                       int n_elements) {
    const int tid    = blockIdx.x * blockDim.x + threadIdx.x;
    const int stride = gridDim.x * blockDim.x;

    const float2* __restrict__ xy = (const float2*)coords;
    const float4* __restrict__ uv = (const float4*)nodal_values;

    float acc = 0.0f;
    for (int e = tid; e < n_elements; e += stride) {
        // Prefetch next iteration's connectivity (streaming input).
        const int pe = e + stride;
        if (pe < n_elements)
            __builtin_prefetch(&elements[3 * pe], 0, 1);

        const int n0 = elements[3 * e + 0];
        const int n1 = elements[3 * e + 1];
        const int n2 = elements[3 * e + 2];

        const float2 p0 = xy[n0];
        const float2 p1 = xy[n1];
        const float2 p2 = xy[n2];
        const float4 u0 = uv[n0];
        const float4 u1 = uv[n1];
        const float4 u2 = uv[n2];

        // J = [[x1-x0, y1-y0],[x2-x0, y2-y0]]   (constant over the element)
        const float ax = p1.x - p0.x, ay = p1.y - p0.y;
        const float bx = p2.x - p0.x, by = p2.y - p0.y;
        const float detJ    = ax * by - ay * bx;
        const float inv_det = 1.0f / detJ;

        // invJ rows; dN_phys rows: dN1 = invJ[0], dN2 = invJ[1], dN0 = -(dN1+dN2)
        const float g1x =  by * inv_det, g1y = -ay * inv_det;
        const float g2x = -bx * inv_det, g2y =  ax * inv_det;
        const float g0x = -(g1x + g2x),  g0y = -(g1y + g2y);

        // u_grad (2 x 4), constant over quads
        float gxv[4], gyv[4];
        const float* U0 = &u0.x; const float* U1 = &u1.x; const float* U2 = &u2.x;
        float grad_sq = 0.0f;
#pragma unroll
        for (int v = 0; v < 4; ++v) {
            gxv[v] = g0x * U0[v] + g1x * U1[v] + g2x * U2[v];
            gyv[v] = g0y * U0[v] + g1y * U1[v] + g2y * U2[v];
            grad_sq += gxv[v] * gxv[v] + gyv[v] * gyv[v];
        }
        const float grad_energy = 0.5f * grad_sq;

        // Shape functions at the 3 quadrature points
        const float NA = 2.0f / 3.0f, NB = 1.0f / 6.0f;
        float val_sum = 0.0f;
#pragma unroll
        for (int q = 0; q < 3; ++q) {
            const float Nq0 = (q == 0) ? NA : NB;
            const float Nq1 = (q == 1) ? NA : NB;
            const float Nq2 = (q == 2) ? NA : NB;
            float s = 0.0f;
#pragma unroll
            for (int v = 0; v < 4; ++v) {
                const float uq = Nq0 * U0[v] + Nq1 * U1[v] + Nq2 * U2[v];
                s += uq * uq;
            }
            val_sum += 0.5f * s;       // val_energy at quad q
        }

        // sum_q w_q*(val_q + grad_energy)*detJ ; w_q = 1/6, sum_q w_q = 1/2
        acc += detJ * ((1.0f / 6.0f) * val_sum + 0.5f * grad_energy);
    }

    const float bs = block_reduce_wmma(acc);
    if (threadIdx.x == 0)
        block_sums[blockIdx.x] = bs;
}

__global__ __launch_bounds__(256)
void final_reduce_kernel(const float* __restrict__ block_sums, int n,
                         float* __restrict__ out) {
    float acc = 0.0f;
    for (int i = threadIdx.x; i < n; i += blockDim.x)
        acc += block_sums[i];
    const float t = block_reduce_wmma(acc);
    if (threadIdx.x == 0)
        out[0] = t;
}

extern "C" void kernel_launch(void* const* d_in, const int* in_sizes, int n_in,
                              void* d_out, int out_size, void* d_ws, size_t ws_size,
                              hipStream_t stream) {
    const float* nodal_values = (const float*)d_in[0];  // (N_NODES, 4) f32
    const float* coords       = (const float*)d_in[1];  // (N_NODES, 2) f32
    const int*   elements     = (const int*)d_in[2];    // (N_ELEM, 3) i32

    const int n_elements = in_sizes[2] / 3;

    int num_blocks = (n_elements + 255) / 256;
    if (num_blocks > 4096) num_blocks = 4096;
    const int ws_cap = (int)(ws_size / sizeof(float));
    if (num_blocks > ws_cap) num_blocks = ws_cap;
    if (num_blocks < 1) num_blocks = 1;

    float* block_sums = (float*)d_ws;

    fem_energy_kernel<<<num_blocks, 256, 0, stream>>>(
        nodal_values, coords, elements, block_sums, n_elements);
    final_reduce_kernel<<<1, 256, 0, stream>>>(
        block_sums, num_blocks, (float*)d_out);
}